// GATRegressor_78469052497932
// MI455X (gfx1250) — compile-verified
//
#include <hip/hip_runtime.h>
#include <hip/hip_bf16.h>

// ---- problem constants (match reference) ----
#define NN   20000
#define TT   4
#define EE   320000
#define IN_F 32
#define H_F  16
#define OUT_F 16
#define HEADS 8
#define ROWS (NN*TT)            // 80000
#define NEG_SLOPE 0.2f

typedef float v2f __attribute__((ext_vector_type(2)));
typedef float v8f __attribute__((ext_vector_type(8)));

// order-preserving float<->uint mapping for atomic max over signed floats
__device__ __forceinline__ unsigned mapf(float x) {
  unsigned u = __float_as_uint(x);
  return u ^ (((int)u < 0) ? 0xFFFFFFFFu : 0x80000000u);
}
__device__ __forceinline__ float unmapf(unsigned u) {
  return __uint_as_float(u ^ (((int)u < 0) ? 0x80000000u : 0xFFFFFFFFu));
}
__device__ __forceinline__ float leaky(float v) {
  return v > 0.0f ? v : NEG_SLOPE * v;
}

// ---------------- init: zero accumulators, mapped(-inf) lower bound = 0, seed out with b2
__global__ __launch_bounds__(256) void init_ws(float* __restrict__ h1,
                                               float* __restrict__ s1, unsigned* __restrict__ m1,
                                               float* __restrict__ s2, unsigned* __restrict__ m2,
                                               float* __restrict__ outp, const float* __restrict__ b2) {
  const int i = blockIdx.x * 256 + threadIdx.x;           // grid covers 10,240,000
  if (i < ROWS * 128) h1[i] = 0.0f;
  if (i < ROWS * HEADS) { s1[i] = 0.0f; m1[i] = 0u; }
  if (i < ROWS)         { s2[i] = 0.0f; m2[i] = 0u; }
  if (i < ROWS * OUT_F) outp[i] = b2[i & 15];
}

// ---------------- layer-1 GEMM: feat1[80000,128] = x[80000,32] @ W1[32,128]  (f32 WMMA)
__global__ __launch_bounds__(128) void gemm1_wmma(const float* __restrict__ x,
                                                  const float* __restrict__ W1,
                                                  float* __restrict__ feat1) {
  const int wave = threadIdx.x >> 5;
  const int lane = threadIdx.x & 31;
  const int half = lane >> 4;        // 0: lanes 0-15, 1: lanes 16-31
  const int m    = lane & 15;
  const int rowTile = blockIdx.x * 4 + wave;              // 5000 tiles exactly
  const int row0 = rowTile << 4;

  // A fragments: lane holds row (row0+m); per K-step of 4, lane half selects K pair
  v2f a[8];
  const float* arow = x + (size_t)(row0 + m) * IN_F + 2 * half;
  #pragma unroll
  for (int i = 0; i < 8; ++i) { a[i].x = arow[i * 4]; a[i].y = arow[i * 4 + 1]; }

  #pragma unroll
  for (int n0 = 0; n0 < 8; ++n0) {                        // 8 column tiles (= heads)
    v8f acc = {0.f,0.f,0.f,0.f,0.f,0.f,0.f,0.f};
    #pragma unroll
    for (int i = 0; i < 8; ++i) {
      const int k = i * 4 + 2 * half;
      v2f b;
      b.x = W1[(size_t)k       * 128 + n0 * 16 + m];
      b.y = W1[(size_t)(k + 1) * 128 + n0 * 16 + m];
      acc = __builtin_amdgcn_wmma_f32_16x16x4_f32(false, a[i], false, b,
                                                  (short)0, acc, false, false);
    }
    float* o = feat1 + (size_t)(row0 + 8 * half) * 128 + n0 * 16 + m;
    #pragma unroll
    for (int r = 0; r < 8; ++r) o[(size_t)r * 128] = acc[r];
  }
}

// ---------------- layer-1 attention scores: el/er[row,h] = <feat1[row,h,:], al/ar[h,:]>
__global__ __launch_bounds__(256) void attn_scores1(const float* __restrict__ feat1,
                                                    const float* __restrict__ al1,
                                                    const float* __restrict__ ar1,
                                                    float* __restrict__ el1, float* __restrict__ er1) {
  const int idx = blockIdx.x * 256 + threadIdx.x;         // 640,000 exactly
  const int row = idx >> 3, h = idx & 7;
  const float* f = feat1 + (size_t)row * 128 + h * 16;
  const float* A = al1 + h * 16;
  const float* B = ar1 + h * 16;
  float el = 0.f, er = 0.f;
  #pragma unroll
  for (int j = 0; j < 16; ++j) { const float v = f[j]; el += v * A[j]; er += v * B[j]; }
  el1[idx] = el; er1[idx] = er;
}

// ---------------- layer-1 segment max (thread per edge*T*H = 10.24M)
__global__ __launch_bounds__(256) void edge_max1(const int* __restrict__ src, const int* __restrict__ dst,
                                                 const float* __restrict__ el1, const float* __restrict__ er1,
                                                 unsigned* __restrict__ m1) {
  const int idx = blockIdx.x * 256 + threadIdx.x;
  const int e = idx >> 5, r = idx & 31;                   // r = t*8+h
  const int s = src[e], d = dst[e];
  const float v = leaky(el1[s * 32 + r] + er1[d * 32 + r]);
  atomicMax(&m1[d * 32 + r], mapf(v));
}

// ---------------- layer-1 segment sum of exp(e - m)
__global__ __launch_bounds__(256) void edge_sum1(const int* __restrict__ src, const int* __restrict__ dst,
                                                 const float* __restrict__ el1, const float* __restrict__ er1,
                                                 const unsigned* __restrict__ m1, float* __restrict__ s1) {
  const int idx = blockIdx.x * 256 + threadIdx.x;
  const int e = idx >> 5, r = idx & 31;
  const int s = src[e], d = dst[e];
  const float v = leaky(el1[s * 32 + r] + er1[d * 32 + r]);
  atomicAdd(&s1[d * 32 + r], expf(v - unmapf(m1[d * 32 + r])));
}

// ---------------- layer-1 aggregation: one wave per (edge,t); lane covers 4 features (head = lane/4)
__global__ __launch_bounds__(256) void edge_agg1(const int* __restrict__ src, const int* __restrict__ dst,
                                                 const float* __restrict__ el1, const float* __restrict__ er1,
                                                 const unsigned* __restrict__ m1, const float* __restrict__ s1,
                                                 const float* __restrict__ feat1, float* __restrict__ h1) {
  const int gid  = (blockIdx.x * 256 + threadIdx.x) >> 5; // wave id over E*T = 1.28M
  const int lane = threadIdx.x & 31;
  const int e = gid >> 2, t = gid & 3;
  const int s = src[e], d = dst[e];
  const int rs = s * 4 + t, rd = d * 4 + t;
  const int h = lane >> 2;
  const float v = leaky(el1[rs * 8 + h] + er1[rd * 8 + h]);
  const float alpha = expf(v - unmapf(m1[rd * 8 + h])) / s1[rd * 8 + h];
  const float4 f = *reinterpret_cast<const float4*>(feat1 + (size_t)rs * 128 + lane * 4);
  float* o = h1 + (size_t)rd * 128 + lane * 4;
  atomicAdd(o + 0, f.x * alpha);
  atomicAdd(o + 1, f.y * alpha);
  atomicAdd(o + 2, f.z * alpha);
  atomicAdd(o + 3, f.w * alpha);
}

// ---------------- layer-2 GEMM: feat2[80000,16] = (h1 + b1)[80000,128] @ W2[128,16]  (f32 WMMA)
__global__ __launch_bounds__(128) void gemm2_wmma(const float* __restrict__ h1,
                                                  const float* __restrict__ b1,
                                                  const float* __restrict__ W2,
                                                  float* __restrict__ feat2) {
  const int wave = threadIdx.x >> 5;
  const int lane = threadIdx.x & 31;
  const int half = lane >> 4;
  const int m    = lane & 15;
  const int rowTile = blockIdx.x * 4 + wave;
  const int row0 = rowTile << 4;

  const float* arow = h1 + (size_t)(row0 + m) * 128 + 2 * half;
  v8f acc = {0.f,0.f,0.f,0.f,0.f,0.f,0.f,0.f};
  #pragma unroll
  for (int i = 0; i < 32; ++i) {
    const int k = i * 4 + 2 * half;
    v2f a, b;
    a.x = arow[i * 4]     + b1[k];
    a.y = arow[i * 4 + 1] + b1[k + 1];
    b.x = W2[(size_t)k       * 16 + m];
    b.y = W2[(size_t)(k + 1) * 16 + m];
    acc = __builtin_amdgcn_wmma_f32_16x16x4_f32(false, a, false, b,
                                                (short)0, acc, false, false);
  }
  float* o = feat2 + (size_t)(row0 + 8 * half) * 16 + m;
  #pragma unroll
  for (int r = 0; r < 8; ++r) o[(size_t)r * 16] = acc[r];
}

// ---------------- layer-2 attention scores (heads=1, 16 features)
__global__ __launch_bounds__(256) void attn_scores2(const float* __restrict__ feat2,
                                                    const float* __restrict__ al2,
                                                    const float* __restrict__ ar2,
                                                    float* __restrict__ el2, float* __restrict__ er2) {
  const int row = blockIdx.x * 256 + threadIdx.x;
  if (row >= ROWS) return;
  const float* f = feat2 + (size_t)row * 16;
  float el = 0.f, er = 0.f;
  #pragma unroll
  for (int j = 0; j < 16; ++j) { const float v = f[j]; el += v * al2[j]; er += v * ar2[j]; }
  el2[row] = el; er2[row] = er;
}

// ---------------- layer-2 segment max / sum (thread per edge*T = 1.28M)
__global__ __launch_bounds__(256) void edge_max2(const int* __restrict__ src, const int* __restrict__ dst,
                                                 const float* __restrict__ el2, const float* __restrict__ er2,
                                                 unsigned* __restrict__ m2) {
  const int idx = blockIdx.x * 256 + threadIdx.x;
  const int e = idx >> 2, t = idx & 3;
  const float v = leaky(el2[src[e] * 4 + t] + er2[dst[e] * 4 + t]);
  atomicMax(&m2[dst[e] * 4 + t], mapf(v));
}

__global__ __launch_bounds__(256) void edge_sum2(const int* __restrict__ src, const int* __restrict__ dst,
                                                 const float* __restrict__ el2, const float* __restrict__ er2,
                                                 const unsigned* __restrict__ m2, float* __restrict__ s2) {
  const int idx = blockIdx.x * 256 + threadIdx.x;
  const int e = idx >> 2, t = idx & 3;
  const int rd = dst[e] * 4 + t;
  const float v = leaky(el2[src[e] * 4 + t] + er2[rd]);
  atomicAdd(&s2[rd], expf(v - unmapf(m2[rd])));
}

// ---------------- layer-2 aggregation into final output (thread per edge*T*16 = 20.48M)
__global__ __launch_bounds__(256) void edge_agg2(const int* __restrict__ src, const int* __restrict__ dst,
                                                 const float* __restrict__ el2, const float* __restrict__ er2,
                                                 const unsigned* __restrict__ m2, const float* __restrict__ s2,
                                                 const float* __restrict__ feat2, float* __restrict__ outp) {
  const int idx = blockIdx.x * 256 + threadIdx.x;
  const int f  = idx & 15;
  const int et = idx >> 4;
  const int e = et >> 2, t = et & 3;
  const int rs = src[e] * 4 + t, rd = dst[e] * 4 + t;
  const float v = leaky(el2[rs] + er2[rd]);
  const float alpha = expf(v - unmapf(m2[rd])) / s2[rd];
  atomicAdd(&outp[(size_t)rd * 16 + f], feat2[(size_t)rs * 16 + f] * alpha);
}

extern "C" void kernel_launch(void* const* d_in, const int* in_sizes, int n_in,
                              void* d_out, int out_size, void* d_ws, size_t ws_size,
                              hipStream_t stream) {
  const float* x   = (const float*)d_in[0];
  const int*   src = (const int*)  d_in[1];
  const int*   dst = (const int*)  d_in[2];
  const float* W1  = (const float*)d_in[3];
  const float* al1 = (const float*)d_in[4];
  const float* ar1 = (const float*)d_in[5];
  const float* b1  = (const float*)d_in[6];
  const float* W2  = (const float*)d_in[7];
  const float* al2 = (const float*)d_in[8];
  const float* ar2 = (const float*)d_in[9];
  const float* b2  = (const float*)d_in[10];
  float* outp = (float*)d_out;

  // workspace layout (floats); total 24,640,000 elems = 98.56 MB
  float* w = (float*)d_ws;
  float*    feat1 = w;               w += (size_t)ROWS * 128;
  float*    h1    = w;               w += (size_t)ROWS * 128;
  float*    el1   = w;               w += (size_t)ROWS * HEADS;
  float*    er1   = w;               w += (size_t)ROWS * HEADS;
  unsigned* m1    = (unsigned*)w;    w += (size_t)ROWS * HEADS;
  float*    s1    = w;               w += (size_t)ROWS * HEADS;
  float*    feat2 = w;               w += (size_t)ROWS * OUT_F;
  float*    el2   = w;               w += (size_t)ROWS;
  float*    er2   = w;               w += (size_t)ROWS;
  unsigned* m2    = (unsigned*)w;    w += (size_t)ROWS;
  float*    s2    = w;

  init_ws    <<<40000,  256, 0, stream>>>(h1, s1, m1, s2, m2, outp, b2);
  gemm1_wmma <<<1250,   128, 0, stream>>>(x, W1, feat1);
  attn_scores1<<<2500,  256, 0, stream>>>(feat1, al1, ar1, el1, er1);
  edge_max1  <<<40000,  256, 0, stream>>>(src, dst, el1, er1, m1);
  edge_sum1  <<<40000,  256, 0, stream>>>(src, dst, el1, er1, m1, s1);
  edge_agg1  <<<160000, 256, 0, stream>>>(src, dst, el1, er1, m1, s1, feat1, h1);
  gemm2_wmma <<<1250,   128, 0, stream>>>(h1, b1, W2, feat2);
  attn_scores2<<<313,   256, 0, stream>>>(feat2, al2, ar2, el2, er2);
  edge_max2  <<<5000,   256, 0, stream>>>(src, dst, el2, er2, m2);
  edge_sum2  <<<5000,   256, 0, stream>>>(src, dst, el2, er2, m2, s2);
  edge_agg2  <<<80000,  256, 0, stream>>>(src, dst, el2, er2, m2, s2, feat2, outp);
}